// ProbabilisticSurfaceDistanceLoss_66400194396562
// MI455X (gfx1250) — compile-verified
//
#include <hip/hip_runtime.h>
#include <stdint.h>

typedef __attribute__((ext_vector_type(2))) float v2f;
typedef __attribute__((ext_vector_type(4))) float v4f;
typedef __attribute__((ext_vector_type(8))) float v8f;

namespace {
constexpr int FS      = 5000;    // simplified faces
constexpr int FS_PAD  = 5008;    // padded to multiple of 16
constexpr int FO      = 30000;   // original faces
constexpr int NVERT   = 20000;   // original vertices
constexpr int NS      = 50;      // samples per face
constexpr int NSAMP   = FS * NS; // 250000
constexpr float PENALTY   = 1e-4f;
constexpr float EPSV      = 1e-8f;
constexpr float REV_SCALE = 0.1f;
constexpr uint32_t BIGU   = 0x7f7fffffu;  // bits of FLT_MAX
}

// ---------------- threefry2x32 (JAX PRNG family), deterministic ----------------
__device__ __forceinline__ uint32_t rotl32(uint32_t x, int r) {
  return (x << r) | (x >> (32 - r));
}

__device__ __forceinline__ void threefry2x32(uint32_t k0, uint32_t k1,
                                             uint32_t c0, uint32_t c1,
                                             uint32_t& o0, uint32_t& o1) {
  const uint32_t ks2 = k0 ^ k1 ^ 0x1BD11BDAu;
  const uint32_t ks[3] = {k0, k1, ks2};
  const int R[8] = {13, 15, 26, 6, 17, 29, 16, 24};
  uint32_t x0 = c0 + k0, x1 = c1 + k1;
#pragma unroll
  for (int b = 0; b < 5; ++b) {
#pragma unroll
    for (int r = 0; r < 4; ++r) {
      x0 += x1;
      x1 = rotl32(x1, R[(b & 1) * 4 + r]);
      x1 ^= x0;
    }
    x0 += ks[(b + 1) % 3];
    x1 += ks[(b + 2) % 3] + (uint32_t)(b + 1);
  }
  o0 = x0; o1 = x1;
}

__device__ __forceinline__ float u01(uint32_t b) {
  return __uint_as_float((b >> 9) | 0x3f800000u) - 1.0f;  // [0,1)
}

// ---------------- prep kernels ----------------
// A-format row: ( x,  y,  z, |p|^2 )   (K=3 A element used by WMMA is 1.0)
// B-format col: (-2x, -2y, -2z, |q|^2) (all four components consumed as-is)

__global__ __launch_bounds__(256) void prep_ob_kernel(
    const float* __restrict__ ov, const int* __restrict__ of,
    float* __restrict__ ob4) {
  int i = blockIdx.x * blockDim.x + threadIdx.x;
  if (i >= FO) return;
  int i0 = of[3 * i], i1 = of[3 * i + 1], i2 = of[3 * i + 2];
  float x = (ov[3 * i0]     + ov[3 * i1]     + ov[3 * i2])     * (1.0f / 3.0f);
  float y = (ov[3 * i0 + 1] + ov[3 * i1 + 1] + ov[3 * i2 + 1]) * (1.0f / 3.0f);
  float z = (ov[3 * i0 + 2] + ov[3 * i1 + 2] + ov[3 * i2 + 2]) * (1.0f / 3.0f);
  float4 o;
  o.x = -2.0f * x; o.y = -2.0f * y; o.z = -2.0f * z;
  o.w = x * x + y * y + z * z;
  ((float4*)ob4)[i] = o;
}

__global__ __launch_bounds__(256) void prep_sb_kernel(
    const float* __restrict__ sv, const int* __restrict__ sf,
    float* __restrict__ sbA) {
  int i = blockIdx.x * blockDim.x + threadIdx.x;
  if (i >= FS_PAD) return;
  float4 o = make_float4(0.f, 0.f, 0.f, 0.f);
  if (i < FS) {
    int i0 = sf[3 * i], i1 = sf[3 * i + 1], i2 = sf[3 * i + 2];
    float x = (sv[3 * i0]     + sv[3 * i1]     + sv[3 * i2])     * (1.0f / 3.0f);
    float y = (sv[3 * i0 + 1] + sv[3 * i1 + 1] + sv[3 * i2 + 1]) * (1.0f / 3.0f);
    float z = (sv[3 * i0 + 2] + sv[3 * i1 + 2] + sv[3 * i2 + 2]) * (1.0f / 3.0f);
    o = make_float4(x, y, z, x * x + y * y + z * z);
  }
  ((float4*)sbA)[i] = o;
}

__global__ __launch_bounds__(256) void prep_tv_kernel(
    const float* __restrict__ ov, float* __restrict__ tv4) {
  int i = blockIdx.x * blockDim.x + threadIdx.x;
  if (i >= NVERT) return;
  float x = ov[3 * i], y = ov[3 * i + 1], z = ov[3 * i + 2];
  float4 o;
  o.x = -2.0f * x; o.y = -2.0f * y; o.z = -2.0f * z;
  o.w = x * x + y * y + z * z;
  ((float4*)tv4)[i] = o;
}

__global__ __launch_bounds__(256) void prep_samples_kernel(
    const float* __restrict__ sv, const int* __restrict__ sf,
    float* __restrict__ sampA) {
  int i = blockIdx.x * blockDim.x + threadIdx.x;
  if (i >= NSAMP) return;
  int f = i / NS;
  int i0 = sf[3 * f], i1 = sf[3 * f + 1], i2 = sf[3 * f + 2];
  uint32_t ra, rb;
  threefry2x32(0x9E3779B9u, 42u, (uint32_t)i, 0x1337BEEFu, ra, rb);
  float sr1 = sqrtf(u01(ra));
  float r2  = u01(rb);
  float a = 1.0f - sr1;
  float b = sr1 * (1.0f - r2);
  float c = sr1 * r2;
  float x = a * sv[3 * i0]     + b * sv[3 * i1]     + c * sv[3 * i2];
  float y = a * sv[3 * i0 + 1] + b * sv[3 * i1 + 1] + c * sv[3 * i2 + 1];
  float z = a * sv[3 * i0 + 2] + b * sv[3 * i1 + 2] + c * sv[3 * i2 + 2];
  ((float4*)sampA)[i] = make_float4(x, y, z, x * x + y * y + z * z);
}

__global__ void init_scal_kernel(float* __restrict__ scal) {
  if (threadIdx.x == 0) { scal[0] = 0.0f; scal[1] = 0.0f; }
}

// ---------------- WMMA distance core: 4 row-tiles (64 rows) per wave ----------------
// D = A(16x4) x B(4x16) + C with A=(x,y,z,1), B=(-2x,-2y,-2z,|q|^2), C[m,*]=|p_m|^2.
// C is loop-invariant per A-tile (separate SRC2/VDST -> no per-iteration movs).
// Min accumulated as v_min_u32 on float bits (valid for non-negative d^2).
__device__ __forceinline__ void wmma_min_tiles4(
    const float* __restrict__ Arows, const int bases[4],
    const float* __restrict__ Bcols, int ncols,
    uint32_t rmin[4][8], int lane) {
  const bool hi = (lane & 16) != 0;
  const int  l15 = lane & 15;

  v2f a[4];
  v8f cini[4];
#pragma unroll
  for (int t = 0; t < 4; ++t) {
    const float* ap = Arows + 4 * (bases[t] + l15);
    float ax = ap[0], ay = ap[1], az = ap[2];
    a[t].x = hi ? az   : ax;   // K=0 | K=2
    a[t].y = hi ? 1.0f : ay;   // K=1 | K=3  (constant 1 in K=3)
    int cb = bases[t] + (hi ? 8 : 0);
#pragma unroll
    for (int r = 0; r < 8; ++r)
      cini[t][r] = Arows[4 * (cb + r) + 3];   // |p_m|^2 for row m = cb + r
  }
#pragma unroll
  for (int t = 0; t < 4; ++t)
#pragma unroll
    for (int r = 0; r < 8; ++r) rmin[t][r] = BIGU;

  for (int n0 = 0; n0 < ncols; n0 += 16) {
    v4f bc = ((const v4f*)Bcols)[n0 + l15];   // one b128, no divergence
    v2f b;
    b.x = hi ? bc.z : bc.x;   // K=0 | K=2
    b.y = hi ? bc.w : bc.y;   // K=1 | K=3  (|q|^2 rides in K=3)
#pragma unroll
    for (int t = 0; t < 4; ++t) {
      v8f d = __builtin_amdgcn_wmma_f32_16x16x4_f32(
          /*neg_a=*/false, a[t], /*neg_b=*/false, b,
          /*c_mod=*/(short)0, cini[t], /*reuse_a=*/false, /*reuse_b=*/false);
#pragma unroll
      for (int r = 0; r < 8; ++r) {
        uint32_t db = __float_as_uint(d[r]);
        rmin[t][r] = rmin[t][r] < db ? rmin[t][r] : db;   // v_min_u32
      }
    }
  }
  // Row-min: reduce across the 16 lanes that hold one row (N dimension).
#pragma unroll
  for (int t = 0; t < 4; ++t)
#pragma unroll
    for (int r = 0; r < 8; ++r) {
      uint32_t v = rmin[t][r];
      uint32_t w;
      w = (uint32_t)__shfl_xor((int)v, 1);  v = v < w ? v : w;
      w = (uint32_t)__shfl_xor((int)v, 2);  v = v < w ? v : w;
      w = (uint32_t)__shfl_xor((int)v, 4);  v = v < w ? v : w;
      w = (uint32_t)__shfl_xor((int)v, 8);  v = v < w ? v : w;
      rmin[t][r] = v;  // uniform within each 16-lane half
    }
}

// ---------------- forward: min over 30000 original barycenters ----------------
__global__ __launch_bounds__(256) void fwd_min_kernel(
    const float* __restrict__ sbA, const float* __restrict__ ob4,
    float* __restrict__ mind) {
  int wave = (blockIdx.x * blockDim.x + threadIdx.x) >> 5;
  int lane = threadIdx.x & 31;
  int m0 = wave * 64;
  if (m0 >= FS_PAD) return;  // wave-uniform: EXEC stays all-ones for WMMA
  int bases[4];
#pragma unroll
  for (int t = 0; t < 4; ++t) {
    int b = m0 + 16 * t;
    bases[t] = b > FS_PAD - 16 ? FS_PAD - 16 : b;  // clamp tail (dup work is benign)
  }
  uint32_t rmin[4][8];
  wmma_min_tiles4(sbA, bases, ob4, FO, rmin, lane);
  if ((lane & 15) == 0) {
    int hi8 = (lane >> 4) << 3;  // lane0 -> rows +0..7, lane16 -> rows +8..15
#pragma unroll
    for (int t = 0; t < 4; ++t)
#pragma unroll
      for (int r = 0; r < 8; ++r) {
        int row = bases[t] + hi8 + r;
        if (row < FS)
          mind[row] = sqrtf(fmaxf(__uint_as_float(rmin[t][r]), 0.0f));
      }
  }
}

// ---------------- reverse: 1-NN of 250000 samples vs 20000 vertices ----------------
__global__ __launch_bounds__(256) void rev_min_kernel(
    const float* __restrict__ sampA, const float* __restrict__ tv4,
    float* __restrict__ dists, float* __restrict__ scal) {
  int wave = (blockIdx.x * blockDim.x + threadIdx.x) >> 5;
  int lane = threadIdx.x & 31;
  int m0 = wave * 64;
  if (m0 >= NSAMP) return;
  int bases[4];
#pragma unroll
  for (int t = 0; t < 4; ++t) {
    int b = m0 + 16 * t;
    bases[t] = b > NSAMP - 16 ? NSAMP - 16 : b;
  }
  uint32_t rmin[4][8];
  wmma_min_tiles4(sampA, bases, tv4, NVERT, rmin, lane);

  float dmax = 0.0f;
  float dloc[4][8];
#pragma unroll
  for (int t = 0; t < 4; ++t)
#pragma unroll
    for (int r = 0; r < 8; ++r) {
      dloc[t][r] = sqrtf(fmaxf(__uint_as_float(rmin[t][r]), 0.0f));
      dmax = fmaxf(dmax, dloc[t][r]);
    }
  if ((lane & 15) == 0) {
    int hi8 = (lane >> 4) << 3;
#pragma unroll
    for (int t = 0; t < 4; ++t)
#pragma unroll
      for (int r = 0; r < 8; ++r)
        dists[bases[t] + hi8 + r] = dloc[t][r];
  }
  // dmax is uniform within each 16-lane half; combine halves, one atomic per wave.
  dmax = fmaxf(dmax, __shfl_xor(dmax, 16));
  if (lane == 0) atomicMax((int*)(scal + 1), __float_as_int(dmax));  // all >= 0
}

// ---------------- reductions ----------------
__global__ __launch_bounds__(256) void fwd_reduce_kernel(
    const float* __restrict__ fp, const float* __restrict__ mind,
    float* __restrict__ scal) {
  __shared__ float sm[256];
  float s = 0.0f;
  for (int i = threadIdx.x; i < FS; i += 256)
    s += fp[i] * mind[i] + PENALTY * (1.0f - fp[i]);
  sm[threadIdx.x] = s;
  __syncthreads();
  for (int w = 128; w > 0; w >>= 1) {
    if (threadIdx.x < w) sm[threadIdx.x] += sm[threadIdx.x + w];
    __syncthreads();
  }
  if (threadIdx.x == 0) scal[0] = sm[0];
}

__global__ __launch_bounds__(256) void final_reduce_kernel(
    const float* __restrict__ fp, const float* __restrict__ dists,
    const float* __restrict__ scal, float* __restrict__ out) {
  __shared__ float sm[256];
  float maxv = scal[1];  // float bits stored via int atomicMax (values >= 0)
  float scale = REV_SCALE / (maxv + EPSV);
  float s = 0.0f;
  for (int i = threadIdx.x; i < NSAMP; i += 256)
    s += fp[i / NS] * dists[i];
  sm[threadIdx.x] = s * scale;
  __syncthreads();
  for (int w = 128; w > 0; w >>= 1) {
    if (threadIdx.x < w) sm[threadIdx.x] += sm[threadIdx.x + w];
    __syncthreads();
  }
  if (threadIdx.x == 0) out[0] = scal[0] + sm[0];
}

// ---------------- launch ----------------
extern "C" void kernel_launch(void* const* d_in, const int* in_sizes, int n_in,
                              void* d_out, int out_size, void* d_ws, size_t ws_size,
                              hipStream_t stream) {
  (void)in_sizes; (void)n_in; (void)out_size; (void)ws_size;
  const float* ov  = (const float*)d_in[0];  // original_vertices  [20000,3]
  const int*   ofc = (const int*)  d_in[1];  // original_faces     [30000,3]
  const float* sv  = (const float*)d_in[2];  // simplified_vertices[4000,3]
  const int*   sfc = (const int*)  d_in[3];  // simplified_faces   [5000,3]
  const float* fp  = (const float*)d_in[4];  // face_probabilities [5000]

  float* ws    = (float*)d_ws;
  float* sbA   = ws;                   // FS_PAD*4   = 20032
  float* ob4   = sbA   + FS_PAD * 4;   // FO*4       = 120000
  float* tv4   = ob4   + FO * 4;       // NVERT*4    = 80000
  float* sampA = tv4   + NVERT * 4;    // NSAMP*4    = 1000000
  float* mind  = sampA + NSAMP * 4;    // FS_PAD     = 5008
  float* dists = mind  + FS_PAD;       // NSAMP      = 250000
  float* scal  = dists + NSAMP;        // 2 scalars  (~5.6 MB total)

  prep_ob_kernel<<<(FO + 255) / 256, 256, 0, stream>>>(ov, ofc, ob4);
  prep_sb_kernel<<<(FS_PAD + 255) / 256, 256, 0, stream>>>(sv, sfc, sbA);
  prep_tv_kernel<<<(NVERT + 255) / 256, 256, 0, stream>>>(ov, tv4);
  prep_samples_kernel<<<(NSAMP + 255) / 256, 256, 0, stream>>>(sv, sfc, sampA);
  init_scal_kernel<<<1, 32, 0, stream>>>(scal);

  // forward: 79 waves of 64 rows each, 8 waves per block
  int fwd_waves  = (FS_PAD + 63) / 64;            // 79
  int fwd_blocks = (fwd_waves + 7) / 8;           // 10
  fwd_min_kernel<<<fwd_blocks, 256, 0, stream>>>(sbA, ob4, mind);
  fwd_reduce_kernel<<<1, 256, 0, stream>>>(fp, mind, scal);

  // reverse: 3907 waves of 64 rows each
  int rev_waves  = (NSAMP + 63) / 64;             // 3907
  int rev_blocks = (rev_waves + 7) / 8;           // 489
  rev_min_kernel<<<rev_blocks, 256, 0, stream>>>(sampA, tv4, dists, scal);
  final_reduce_kernel<<<1, 256, 0, stream>>>(fp, dists, scal, (float*)d_out);
}